// Updater_45595372814771
// MI455X (gfx1250) — compile-verified
//
#include <hip/hip_runtime.h>
#include <hip/hip_bf16.h>

// Problem constants (match reference)
#define N_NODES 50000
#define N_EDGES 800000
#define D_FEAT  64
#define BATCH   2

// Tiling: 256-thread blocks = 8 wave32s. Each wave processes 32-edge groups,
// double-buffering edge metadata in LDS via gfx1250 async global->LDS copies.
#define WAVES_PER_BLOCK 8
#define GROUP_EDGES     32
#define BLOCK_EDGES     (WAVES_PER_BLOCK * GROUP_EDGES)   // 256
// Per-wave per-buffer LDS layout (512 bytes):
//   [  0..127]  int32 src[32]
//   [128..255]  int32 dst[32]
//   [256..511]  float2 weight[32]   (weight layout is [E][B], B=2 -> contiguous pairs)
#define BUF_BYTES 512
#define LDS_BYTES (WAVES_PER_BLOCK * 2 * BUF_BYTES)       // 8 KB

// ---------------------------------------------------------------------------
// No-return fp32 atomic add at device scope: RMW performed by the L2 atomic
// units, tracked with STOREcnt (no wait needed before S_ENDPGM's implicit
// wait-idle). Guarantees global_atomic_add_f32 (no CAS loop).
__device__ __forceinline__ void atom_add_f32(const float* base, unsigned byte_off, float v) {
    asm volatile("global_atomic_add_f32 %0, %1, %2 scope:SCOPE_DEV"
                 :: "v"(byte_off), "v"(v), "s"(base)
                 : "memory");
}

// Async global->LDS copies (GVS mode: SADDR base + 32-bit VGPR byte offset).
// Tracked with ASYNCcnt; completion = data visible in LDS.
__device__ __forceinline__ void async_b32_to_lds(const void* base, unsigned goff, unsigned lds_off) {
    asm volatile("global_load_async_to_lds_b32 %0, %1, %2"
                 :: "v"(lds_off), "v"(goff), "s"(base)
                 : "memory");
}
__device__ __forceinline__ void async_b64_to_lds(const void* base, unsigned goff, unsigned lds_off) {
    asm volatile("global_load_async_to_lds_b64 %0, %1, %2"
                 :: "v"(lds_off), "v"(goff), "s"(base)
                 : "memory");
}

// Issue the 3 async copies for one 32-edge group (this wave's own buffer).
// ebase is a multiple of 32 and E % 32 == 0, so the group is always full.
__device__ __forceinline__ void issue_group(const int* ei, const float* wgt,
                                            unsigned bufbase, int lane, long long ebase) {
    unsigned e = (unsigned)(ebase + lane);
    async_b32_to_lds(ei,  e * 4u,                              bufbase + (unsigned)lane * 4u);
    async_b32_to_lds(ei,  (unsigned)N_EDGES * 4u + e * 4u,     bufbase + 128u + (unsigned)lane * 4u);
    async_b64_to_lds(wgt, e * 8u,                              bufbase + 256u + (unsigned)lane * 8u);
}

// ---------------------------------------------------------------------------
__global__ void __launch_bounds__(256)
zero_out_kernel(float4* __restrict__ out4, int n4) {
    int i = blockIdx.x * blockDim.x + threadIdx.x;
    int stride = gridDim.x * blockDim.x;
    for (; i < n4; i += stride) out4[i] = float4{0.f, 0.f, 0.f, 0.f};
}

__global__ void __launch_bounds__(256)
scatter_add_kernel(const float* __restrict__ state,   // [B, N, D]
                   const int*   __restrict__ ei,      // [2, E] (src row, dst row)
                   const float* __restrict__ wgt,     // [E, B]
                   float*       __restrict__ out) {   // [B, N, D]
    extern __shared__ unsigned char dynlds[];         // only LDS user -> offset 0
    const int lane = threadIdx.x & 31;
    const int wave = threadIdx.x >> 5;

    const long long step  = (long long)gridDim.x * BLOCK_EDGES;           // edges per grid sweep
    const long long base0 = (long long)blockIdx.x * BLOCK_EDGES + (long long)wave * GROUP_EDGES;
    const unsigned bb0 = (unsigned)((wave * 2 + 0) * BUF_BYTES);
    const unsigned bb1 = (unsigned)((wave * 2 + 1) * BUF_BYTES);

    const float* state1 = state + (size_t)N_NODES * D_FEAT;               // batch-1 slab
    float*       out1   = out   + (size_t)N_NODES * D_FEAT;

    // Prologue: fill both buffers of the double buffer.
    if (base0 < (long long)N_EDGES)        issue_group(ei, wgt, bb0, lane, base0);
    if (base0 + step < (long long)N_EDGES) issue_group(ei, wgt, bb1, lane, base0 + step);

    unsigned cur_bb = bb0, oth_bb = bb1;
    for (long long base = base0; base < (long long)N_EDGES; base += step) {
        // 3 async ops per group; in-order completion => waiting to <=3 retires
        // the current buffer while the other buffer's 3 ops stay in flight.
        if (base + step < (long long)N_EDGES)
            asm volatile("s_wait_asynccnt 3" ::: "memory");
        else
            asm volatile("s_wait_asynccnt 0" ::: "memory");

        const int*    srcA = (const int*)   (dynlds + cur_bb);
        const int*    dstA = (const int*)   (dynlds + cur_bb + 128);
        const float2* wA   = (const float2*)(dynlds + cur_bb + 256);

        #pragma unroll 4
        for (int i = 0; i < GROUP_EDGES; ++i) {
            const int    s  = srcA[i];          // LDS broadcast reads (wave-uniform)
            const int    d  = dstA[i];
            const float2 wv = wA[i];            // wv.x = weight[e][0], wv.y = weight[e][1]

            // Coalesced gather: 32 lanes x float2 = full 256 B row per batch.
            const float2 a0 = ((const float2*)state )[(size_t)s * 32 + lane];
            const float2 a1 = ((const float2*)state1)[(size_t)s * 32 + lane];

            // Scatter-add: byte offset of (row d, feature 2*lane) within a batch slab.
            const unsigned od = (unsigned)d * (D_FEAT * 4u) + (unsigned)lane * 8u;
            atom_add_f32(out,  od,      a0.x * wv.x);
            atom_add_f32(out,  od + 4u, a0.y * wv.x);
            atom_add_f32(out1, od,      a1.x * wv.y);
            atom_add_f32(out1, od + 4u, a1.y * wv.y);
        }

        // Refill the buffer we just drained with the group two steps ahead.
        const long long nxt2 = base + 2 * step;
        if (nxt2 < (long long)N_EDGES) issue_group(ei, wgt, cur_bb, lane, nxt2);
        const unsigned t = cur_bb; cur_bb = oth_bb; oth_bb = t;
    }
}

// ---------------------------------------------------------------------------
extern "C" void kernel_launch(void* const* d_in, const int* in_sizes, int n_in,
                              void* d_out, int out_size, void* d_ws, size_t ws_size,
                              hipStream_t stream) {
    const float* state = (const float*)d_in[0];   // [B, N, D] fp32
    const int*   ei    = (const int*)  d_in[1];   // [2, E] int
    const float* wgt   = (const float*)d_in[2];   // [E, B] fp32
    float*       out   = (float*)d_out;           // [B, N, D] fp32

    (void)in_sizes; (void)n_in; (void)d_ws; (void)ws_size;

    // 1) Zero the output (harness poisons d_out; we accumulate with atomics).
    const int n4 = (BATCH * N_NODES * D_FEAT) / 4;            // 1.6M float4
    const int zgrid = (n4 + 255) / 256;
    zero_out_kernel<<<zgrid, 256, 0, stream>>>((float4*)out, n4);

    // 2) Edge-parallel gather/scale/scatter. 1024 blocks x 8 waves = 8192 wave
    //    streams over 25000 edge-groups (~3 groups/wave -> double buffer active).
    scatter_add_kernel<<<1024, 256, LDS_BYTES, stream>>>(state, ei, wgt, out);
}